// EntropicLayer_81509889343814
// MI455X (gfx1250) — compile-verified
//
#include <hip/hip_runtime.h>
#include <hip/hip_bf16.h>
#include <float.h>
#include <math.h>

#define N_NODES 50000
#define N_EDGES 640000
#define D 128

typedef float v2f __attribute__((ext_vector_type(2)));
typedef float v8f __attribute__((ext_vector_type(8)));

// ---------------------------------------------------------------- init
__global__ __launch_bounds__(256) void k_init(float* deg, float* Ebuf, unsigned int* red) {
    int i = blockIdx.x * 256 + threadIdx.x;
    if (i < N_NODES) { deg[i] = 1.0f; Ebuf[i] = 0.0f; }   // deg starts at 1 (self-loop)
    if (i == 0) {
        red[0] = 0x7F7FFFFFu;          // FLT_MAX bits: min-E accumulator (E >= 0)
        ((float*)red)[1] = 0.0f;       // sum exp
        ((float*)red)[2] = 0.0f;       // sum p*logp
    }
}

// ---------------------------------------------------------------- in-degree
__global__ __launch_bounds__(256) void k_deg(const long long* __restrict__ ei, float* deg) {
    int e = blockIdx.x * 256 + threadIdx.x;
    if (e < N_EDGES) atomicAdd(&deg[(int)ei[N_EDGES + e]], 1.0f);
}

// ---------------------------------------------------------------- y = x / deg  (self-loop term)
__global__ __launch_bounds__(256) void k_self(const float* __restrict__ x,
                                              const float* __restrict__ deg,
                                              float* __restrict__ y) {
    int i4 = blockIdx.x * 256 + threadIdx.x;
    if (i4 >= N_NODES * D / 4) return;
    int base = i4 * 4;
    int n = base >> 7;
    float inv = 1.0f / deg[n];
    float4 xv = *(const float4*)(x + base);
    float4 o; o.x = xv.x * inv; o.y = xv.y * inv; o.z = xv.z * inv; o.w = xv.w * inv;
    *(float4*)(y + base) = o;
}

// ---------------------------------------------------------------- y[d] += norm * x[s]  (one wave per edge)
__global__ __launch_bounds__(256) void k_agg(const float* __restrict__ x,
                                             const long long* __restrict__ ei,
                                             const float* __restrict__ deg,
                                             float* __restrict__ y) {
    int e = blockIdx.x * 8 + (threadIdx.x >> 5);
    if (e >= N_EDGES) return;
    int lane = threadIdx.x & 31;
    int s = (int)ei[e];
    int d = (int)ei[N_EDGES + e];
    float norm = rsqrtf(deg[s] * deg[d]);
    float4 xv = *(const float4*)(x + (size_t)s * D + lane * 4);
    float* yp = y + (size_t)d * D + lane * 4;
    atomicAdd(yp + 0, norm * xv.x);
    atomicAdd(yp + 1, norm * xv.y);
    atomicAdd(yp + 2, norm * xv.z);
    atomicAdd(yp + 3, norm * xv.w);
}

// ---------------------------------------------------------------- H = Y @ W + b  via V_WMMA_F32_16X16X4_F32
// Block = 256 threads = 8 waves. Block covers 16 rows; wave w covers cols [16w,16w+16).
__global__ __launch_bounds__(256) void k_gemm(const float* __restrict__ Y,
                                              const float* __restrict__ Wm,
                                              const float* __restrict__ bias,
                                              float* __restrict__ H) {
    __shared__ float sW[32][132];   // 32 k-rows x 128 cols, padded (pitch 132 -> conflict-free)
    __shared__ float sY[16][36];    // 16 rows x 32 k, padded
    const int tid   = threadIdx.x;
    const int wave  = tid >> 5;
    const int lane  = tid & 31;
    const int m     = lane & 15;
    const int khalf = lane >> 4;          // 0: k∈{0,1}, 1: k∈{2,3} within each K=4 step
    const int row0  = blockIdx.x * 16;
    const int col   = wave * 16 + m;

    v8f acc = {0.f,0.f,0.f,0.f,0.f,0.f,0.f,0.f};

    for (int kc = 0; kc < D; kc += 32) {
        // stage 32x128 slab of W
        for (int i = tid; i < 32 * 32; i += 256) {       // 1024 float4s
            int r  = i >> 5;
            int c4 = (i & 31) << 2;
            float4 wv = *(const float4*)(&Wm[(kc + r) * D + c4]);
            sW[r][c4 + 0] = wv.x; sW[r][c4 + 1] = wv.y;
            sW[r][c4 + 2] = wv.z; sW[r][c4 + 3] = wv.w;
        }
        // stage 16x32 tile of Y
        if (tid < 128) {
            int r  = tid >> 3;
            int c4 = (tid & 7) << 2;
            float4 yv = *(const float4*)(&Y[(size_t)(row0 + r) * D + kc + c4]);
            sY[r][c4 + 0] = yv.x; sY[r][c4 + 1] = yv.y;
            sY[r][c4 + 2] = yv.z; sY[r][c4 + 3] = yv.w;
        }
        __syncthreads();
        #pragma unroll
        for (int kt = 0; kt < 8; ++kt) {
            int kl = kt * 4 + khalf * 2;
            v2f a; a.x = sY[m][kl];            a.y = sY[m][kl + 1];
            v2f b; b.x = sW[kl][col];          b.y = sW[kl + 1][col];
#if __has_builtin(__builtin_amdgcn_wmma_f32_16x16x4_f32)
            acc = __builtin_amdgcn_wmma_f32_16x16x4_f32(
                      false, a, false, b, (short)0, acc, false, false);
#else
            // fallback (should not be taken on gfx1250): per-lane FMA, layout-compatible only as placeholder
            acc[0] += a.x * b.x + a.y * b.y;
#endif
        }
        __syncthreads();
    }
    // C/D layout: VGPR v -> row (v + 8*khalf), lane%16 -> col
    float bb = bias[col];
    #pragma unroll
    for (int v = 0; v < 8; ++v) {
        int row = row0 + v + 8 * khalf;
        H[(size_t)row * D + col] = acc[v] + bb;
    }
}

// ---------------------------------------------------------------- zero d_out (grad accumulator)
__global__ __launch_bounds__(256) void k_zero(float* __restrict__ g) {
    int i4 = blockIdx.x * 256 + threadIdx.x;
    if (i4 >= N_NODES * D / 4) return;
    float4 z; z.x = 0.f; z.y = 0.f; z.z = 0.f; z.w = 0.f;
    *(float4*)(g + i4 * 4) = z;
}

// ---------------------------------------------------------------- E[dst] += ||h_s - h_d||^2  (wave per edge)
__global__ __launch_bounds__(256) void k_energy(const float* __restrict__ H,
                                                const long long* __restrict__ ei,
                                                float* __restrict__ Ebuf) {
    int e = blockIdx.x * 8 + (threadIdx.x >> 5);
    if (e >= N_EDGES) return;
    int lane = threadIdx.x & 31;
    int s = (int)ei[e];
    int d = (int)ei[N_EDGES + e];
    float4 hs = *(const float4*)(H + (size_t)s * D + lane * 4);
    float4 hd = *(const float4*)(H + (size_t)d * D + lane * 4);
    float dx = hs.x - hd.x, dy = hs.y - hd.y, dz = hs.z - hd.z, dw = hs.w - hd.w;
    float part = dx * dx + dy * dy + dz * dz + dw * dw;
    #pragma unroll
    for (int off = 16; off > 0; off >>= 1)
        part += __shfl_xor(part, off);
    if (lane == 0) atomicAdd(&Ebuf[d], part);
}

// ---------------------------------------------------------------- min(E) via uint atomicMin (E >= 0)
__global__ __launch_bounds__(256) void k_redmin(const float* __restrict__ Ebuf, unsigned int* red) {
    __shared__ float sm[256];
    int i = blockIdx.x * 256 + threadIdx.x;
    sm[threadIdx.x] = (i < N_NODES) ? Ebuf[i] : FLT_MAX;
    __syncthreads();
    for (int s = 128; s > 0; s >>= 1) {
        if (threadIdx.x < s) sm[threadIdx.x] = fminf(sm[threadIdx.x], sm[threadIdx.x + s]);
        __syncthreads();
    }
    if (threadIdx.x == 0) atomicMin(&red[0], __float_as_uint(sm[0]));
}

// ---------------------------------------------------------------- S = sum exp((minE - E)/T)
__global__ __launch_bounds__(256) void k_sumexp(const float* __restrict__ Ebuf,
                                                const float* __restrict__ temp,
                                                unsigned int* red) {
    __shared__ float sm[256];
    const float* redf = (const float*)red;
    float minE = redf[0];
    float invT = 1.0f / temp[0];
    int i = blockIdx.x * 256 + threadIdx.x;
    sm[threadIdx.x] = (i < N_NODES) ? expf((minE - Ebuf[i]) * invT) : 0.0f;
    __syncthreads();
    for (int s = 128; s > 0; s >>= 1) {
        if (threadIdx.x < s) sm[threadIdx.x] += sm[threadIdx.x + s];
        __syncthreads();
    }
    if (threadIdx.x == 0) atomicAdd(&((float*)red)[1], sm[0]);
}

// ---------------------------------------------------------------- sum p*logp
__global__ __launch_bounds__(256) void k_plogp(const float* __restrict__ Ebuf,
                                               const float* __restrict__ temp,
                                               unsigned int* red) {
    __shared__ float sm[256];
    const float* redf = (const float*)red;
    float minE = redf[0];
    float logS = logf(redf[1]);
    float invT = 1.0f / temp[0];
    int i = blockIdx.x * 256 + threadIdx.x;
    float v = 0.0f;
    if (i < N_NODES) {
        float lp = (minE - Ebuf[i]) * invT - logS;
        v = expf(lp) * lp;
    }
    sm[threadIdx.x] = v;
    __syncthreads();
    for (int s = 128; s > 0; s >>= 1) {
        if (threadIdx.x < s) sm[threadIdx.x] += sm[threadIdx.x + s];
        __syncthreads();
    }
    if (threadIdx.x == 0) atomicAdd(&((float*)red)[2], sm[0]);
}

// ---------------------------------------------------------------- c_k = p_k (logp_k + H) / T
__global__ __launch_bounds__(256) void k_coef(const float* __restrict__ Ebuf,
                                              const float* __restrict__ temp,
                                              const unsigned int* red,
                                              float* __restrict__ Cbuf) {
    int i = blockIdx.x * 256 + threadIdx.x;
    if (i >= N_NODES) return;
    const float* redf = (const float*)red;
    float minE = redf[0];
    float logS = logf(redf[1]);
    float Hent = -redf[2];
    float invT = 1.0f / temp[0];
    float lp = (minE - Ebuf[i]) * invT - logS;
    Cbuf[i] = expf(lp) * (lp + Hent) * invT;
}

// ---------------------------------------------------------------- grad scatter: t = 2 c_d (h_s - h_d); g_s += t; g_d -= t
__global__ __launch_bounds__(256) void k_grad(const float* __restrict__ H,
                                              const long long* __restrict__ ei,
                                              const float* __restrict__ Cbuf,
                                              float* __restrict__ g) {
    int e = blockIdx.x * 8 + (threadIdx.x >> 5);
    if (e >= N_EDGES) return;
    int lane = threadIdx.x & 31;
    int s = (int)ei[e];
    int d = (int)ei[N_EDGES + e];
    float cd = 2.0f * Cbuf[d];
    float4 hs = *(const float4*)(H + (size_t)s * D + lane * 4);
    float4 hd = *(const float4*)(H + (size_t)d * D + lane * 4);
    float tx = cd * (hs.x - hd.x), ty = cd * (hs.y - hd.y);
    float tz = cd * (hs.z - hd.z), tw = cd * (hs.w - hd.w);
    float* gs = g + (size_t)s * D + lane * 4;
    float* gd = g + (size_t)d * D + lane * 4;
    atomicAdd(gs + 0,  tx); atomicAdd(gs + 1,  ty);
    atomicAdd(gs + 2,  tz); atomicAdd(gs + 3,  tw);
    atomicAdd(gd + 0, -tx); atomicAdd(gd + 1, -ty);
    atomicAdd(gd + 2, -tz); atomicAdd(gd + 3, -tw);
}

// ---------------------------------------------------------------- out = h + w * g (g lives in d_out)
__global__ __launch_bounds__(256) void k_final(const float* __restrict__ H,
                                               const float* __restrict__ weight,
                                               float* __restrict__ out) {
    int i4 = blockIdx.x * 256 + threadIdx.x;
    if (i4 >= N_NODES * D / 4) return;
    float w = weight[0];
    int base = i4 * 4;
    float4 h = *(const float4*)(H + base);
    float4 gv = *(const float4*)(out + base);
    float4 o;
    o.x = h.x + w * gv.x; o.y = h.y + w * gv.y;
    o.z = h.z + w * gv.z; o.w = h.w + w * gv.w;
    *(float4*)(out + base) = o;
}

// ================================================================ launcher
extern "C" void kernel_launch(void* const* d_in, const int* in_sizes, int n_in,
                              void* d_out, int out_size, void* d_ws, size_t ws_size,
                              hipStream_t stream) {
    const float*     x    = (const float*)d_in[0];
    const long long* ei   = (const long long*)d_in[1];   // int64 edge_index [2, E]
    const float*     wgt  = (const float*)d_in[2];
    const float*     temp = (const float*)d_in[3];
    const float*     Wm   = (const float*)d_in[4];
    const float*     bias = (const float*)d_in[5];
    float*           out  = (float*)d_out;

    // workspace layout (floats): H[N*D] | deg[N] | E[N] | C[N] | red[4]
    float* H    = (float*)d_ws;
    float* deg  = H + (size_t)N_NODES * D;
    float* Ebuf = deg + N_NODES;
    float* Cbuf = Ebuf + N_NODES;
    unsigned int* red = (unsigned int*)(Cbuf + N_NODES);

    const int nBlkN   = (N_NODES + 255) / 256;
    const int nBlkE   = (N_EDGES + 255) / 256;
    const int nBlkEW  = N_EDGES / 8;            // wave-per-edge kernels (exact)
    const int nBlkV4  = (N_NODES * D / 4) / 256; // exact: 6250
    const int nBlkGemm = N_NODES / 16;           // exact: 3125

    k_init  <<<nBlkN,   256, 0, stream>>>(deg, Ebuf, red);
    k_deg   <<<nBlkE,   256, 0, stream>>>(ei, deg);
    k_self  <<<nBlkV4,  256, 0, stream>>>(x, deg, out);          // out := x/deg (self-loop)
    k_agg   <<<nBlkEW,  256, 0, stream>>>(x, ei, deg, out);      // out += norm * x[src]
    k_gemm  <<<nBlkGemm,256, 0, stream>>>(out, Wm, bias, H);     // H = out @ W + b
    k_zero  <<<nBlkV4,  256, 0, stream>>>(out);                  // out := 0 (grad accum)
    k_energy<<<nBlkEW,  256, 0, stream>>>(H, ei, Ebuf);
    k_redmin<<<nBlkN,   256, 0, stream>>>(Ebuf, red);
    k_sumexp<<<nBlkN,   256, 0, stream>>>(Ebuf, temp, red);
    k_plogp <<<nBlkN,   256, 0, stream>>>(Ebuf, temp, red);
    k_coef  <<<nBlkN,   256, 0, stream>>>(Ebuf, temp, red, Cbuf);
    k_grad  <<<nBlkEW,  256, 0, stream>>>(H, ei, Cbuf, out);
    k_final <<<nBlkV4,  256, 0, stream>>>(H, wgt, out);
}